// InstanceSegmentationLoss_81801947120084
// MI455X (gfx1250) — compile-verified
//
#include <hip/hip_runtime.h>
#include <cstdint>

// Instance-segmentation loss: joint-histogram reformulation of the reference's
// one-hot matmul. Memory floor ~8MB / 23.3TB/s ~= 0.36us; histogram is the
// bandwidth-optimal algorithm on MI455X. WMMA (f32 16x16x4) is used where it
// is exact and natural: row/col sums of the count matrix (areas).

#define NUM_INST 64
#define NBIN     65                  // ids 0..64 (0 = background)
#define HSIZE    (NBIN * NBIN)       // 4225 bins
#define PITCH    68                  // padded K (65 -> 68, multiple of 4)
#define PSIZE    (PITCH * PITCH)     // zero-padded LDS matrix
#define HBLOCK   256
#define HGRID    256

typedef float v2f __attribute__((ext_vector_type(2)));
typedef float v8f __attribute__((ext_vector_type(8)));
typedef int   v4i __attribute__((vector_size(16)));   // matches builtin's V4i

#if __has_builtin(__builtin_amdgcn_global_load_async_to_lds_b128) && \
    __has_builtin(__builtin_amdgcn_s_wait_asynccnt)
#define USE_ASYNC_LDS 1
#else
#define USE_ASYNC_LDS 0
#endif

// AS-qualified vector pointers for the async-LDS builtin (AS1 = global src,
// AS3 = LDS dst); C-style cast performs the addrspacecast.
#define AS1V(p) ((__attribute__((address_space(1))) v4i*)(p))
#define AS3V(p) ((__attribute__((address_space(3))) v4i*)(p))

// ---------------------------------------------------------------- init ------
__global__ __launch_bounds__(256) void iseg_init_hist(uint32_t* __restrict__ g_hist) {
  int i = blockIdx.x * blockDim.x + threadIdx.x;
  if (i < HSIZE) g_hist[i] = 0u;
}

// ----------------------------------------------------------- histogram ------
__global__ __launch_bounds__(HBLOCK) void iseg_hist(const int* __restrict__ pred,
                                                    const int* __restrict__ gt,
                                                    uint32_t* __restrict__ g_hist,
                                                    int n4) {
  __shared__ uint32_t lh[HSIZE];
#if USE_ASYNC_LDS
  __shared__ int4 stP[2][HBLOCK];
  __shared__ int4 stT[2][HBLOCK];
#endif
  const int tid = threadIdx.x;
  for (int i = tid; i < HSIZE; i += HBLOCK) lh[i] = 0u;
  __syncthreads();

  const int4* p4 = reinterpret_cast<const int4*>(pred);
  const int4* t4 = reinterpret_cast<const int4*>(gt);
  const int stride = HBLOCK * HGRID;
  const int i0 = blockIdx.x * HBLOCK + tid;

#if USE_ASYNC_LDS
  // Double-buffered async global->LDS staging. Each lane owns its private
  // LDS slots, so only per-wave ASYNCcnt waits are needed (no barriers).
  if (i0 < n4) {
    __builtin_amdgcn_global_load_async_to_lds_b128(AS1V(p4 + i0), AS3V(&stP[0][tid]), 0, 0);
    __builtin_amdgcn_global_load_async_to_lds_b128(AS1V(t4 + i0), AS3V(&stT[0][tid]), 0, 0);
  }
  int buf = 0;
  for (int i = i0; i < n4; i += stride) {
    const int nx = i + stride;
    if (nx < n4) {
      __builtin_amdgcn_global_load_async_to_lds_b128(AS1V(p4 + nx), AS3V(&stP[buf ^ 1][tid]), 0, 0);
      __builtin_amdgcn_global_load_async_to_lds_b128(AS1V(t4 + nx), AS3V(&stT[buf ^ 1][tid]), 0, 0);
      __builtin_amdgcn_s_wait_asynccnt(2);   // current buffer's 2 loads done (in-order)
    } else {
      __builtin_amdgcn_s_wait_asynccnt(0);
    }
    const int4 p = stP[buf][tid];
    const int4 t = stT[buf][tid];
    atomicAdd(&lh[p.x * NBIN + t.x], 1u);
    atomicAdd(&lh[p.y * NBIN + t.y], 1u);
    atomicAdd(&lh[p.z * NBIN + t.z], 1u);
    atomicAdd(&lh[p.w * NBIN + t.w], 1u);
    buf ^= 1;
  }
#else
  for (int i = i0; i < n4; i += stride) {
    const int4 p = p4[i];
    const int4 t = t4[i];
    atomicAdd(&lh[p.x * NBIN + t.x], 1u);
    atomicAdd(&lh[p.y * NBIN + t.y], 1u);
    atomicAdd(&lh[p.z * NBIN + t.z], 1u);
    atomicAdd(&lh[p.w * NBIN + t.w], 1u);
  }
#endif

  __syncthreads();
  // Skip-zero merge into the global histogram (integer adds: deterministic).
  for (int i = tid; i < HSIZE; i += HBLOCK) {
    const uint32_t v = lh[i];
    if (v) atomicAdd(&g_hist[i], v);
  }
}

// ------------------------------------------------------------ finalize ------
__global__ __launch_bounds__(32) void iseg_finalize(const uint32_t* __restrict__ g_hist,
                                                    float* __restrict__ out) {
  // Counts stored zero-padded at 68x68 so all WMMA gathers (K padded to 68)
  // are unconditionally in-bounds -> straight-line ds_load + v_wmma code,
  // no divergent EXEC masking around the matrix ops.
  __shared__ float cF[PSIZE];
  __shared__ float areaP[NUM_INST];
  __shared__ float areaT[NUM_INST];
  __shared__ float red[64];

  const int lane = threadIdx.x;
  for (int i = lane; i < PSIZE; i += 32) cF[i] = 0.0f;
  __syncthreads();
  for (int i = lane; i < HSIZE; i += 32) {
    const int r = i / NBIN;
    const int c = i - r * NBIN;
    cF[r * PITCH + c] = (float)g_hist[i];   // exact: counts < 2^24
  }
  __syncthreads();

#if __has_builtin(__builtin_amdgcn_wmma_f32_16x16x4_f32)
  // area_p[p] = sum_t c[p][t], area_t[t] = sum_p c[p][t] as 16x16x4 f32 WMMA
  // against a ones B-matrix (B layout irrelevant: all entries are 1.0).
  // A layout (ISA 7.12.2, 32-bit 16x4): lanes 0-15 hold K={0,1}, lanes 16-31 K={2,3}.
  const v2f onesB = {1.0f, 1.0f};
  const int m = lane & 15;
  const int halfSel = lane >> 4;
  for (int rb = 0; rb < NUM_INST; rb += 16) {
    v8f accP = {0.f, 0.f, 0.f, 0.f, 0.f, 0.f, 0.f, 0.f};
    v8f accT = {0.f, 0.f, 0.f, 0.f, 0.f, 0.f, 0.f, 0.f};
    const int row = rb + m;               // id index 0..63 -> matrix row id row+1
    for (int k0 = 0; k0 < PITCH; k0 += 4) {  // K = 65, zero-padded to 68
      const int kA = k0 + 2 * halfSel;
      v2f A;  // A[m][k] = c[row+1][k]  (sum over t incl. background col 0)
      A.x = cF[(row + 1) * PITCH + kA];
      A.y = cF[(row + 1) * PITCH + kA + 1];
      accP = __builtin_amdgcn_wmma_f32_16x16x4_f32(false, A, false, onesB,
                                                   (short)0, accP, false, false);
      v2f A2; // A2[m][k] = c[k][row+1] (transposed gather -> column sums)
      A2.x = cF[kA * PITCH + (row + 1)];
      A2.y = cF[(kA + 1) * PITCH + (row + 1)];
      accT = __builtin_amdgcn_wmma_f32_16x16x4_f32(false, A2, false, onesB,
                                                   (short)0, accT, false, false);
    }
    // D layout: VGPR r -> (lanes 0-15: M=r, N=lane), (lanes 16-31: M=r+8).
    // Every column of D is identical; extract column N=0 from lanes 0 and 16.
    if (lane == 0) {
#pragma unroll
      for (int r = 0; r < 8; ++r) { areaP[rb + r] = accP[r]; areaT[rb + r] = accT[r]; }
    }
    if (lane == 16) {
#pragma unroll
      for (int r = 0; r < 8; ++r) { areaP[rb + 8 + r] = accP[r]; areaT[rb + 8 + r] = accT[r]; }
    }
  }
#else
  for (int i = lane; i < NUM_INST; i += 32) {
    float sp = 0.0f, st = 0.0f;
    for (int k = 0; k < NBIN; ++k) {
      sp += cF[(i + 1) * PITCH + k];
      st += cF[k * PITCH + (i + 1)];
    }
    areaP[i] = sp;
    areaT[i] = st;
  }
#endif
  __syncthreads();

  float lossSum = 0.0f, num = 0.0f;
  for (int i = lane; i < NUM_INST; i += 32) {   // i = lane, lane+32
    const float ap = areaP[i];
    const float at = areaT[i];
    float maxP = 0.0f, maxT = 0.0f;
    for (int j = 0; j < NUM_INST; ++j) {
      // pred instance i+1 vs true instance j+1
      const float iP = cF[(i + 1) * PITCH + (j + 1)];
      const float uP = ap + areaT[j] - iP;
      maxP = fmaxf(maxP, (uP > 0.0f) ? iP / fmaxf(uP, 1.0f) : 0.0f);
      // true instance i+1 vs pred instance j+1
      const float iT = cF[(j + 1) * PITCH + (i + 1)];
      const float uT = areaP[j] + at - iT;
      maxT = fmaxf(maxT, (uT > 0.0f) ? iT / fmaxf(uT, 1.0f) : 0.0f);
    }
    if (ap > 0.0f) { lossSum += 1.0f - maxP; num += 1.0f; }
    if (at > 0.0f) { lossSum += 1.0f - maxT; num += 1.0f; }
  }
  red[lane] = lossSum;
  red[32 + lane] = num;
  __syncthreads();
  if (lane == 0) {
    float ls = 0.0f, n = 0.0f;
    for (int l = 0; l < 32; ++l) { ls += red[l]; n += red[32 + l]; }
    out[0] = (n > 0.0f) ? ls / n : 0.0f;
  }
}

// -------------------------------------------------------------- launch ------
extern "C" void kernel_launch(void* const* d_in, const int* in_sizes, int n_in,
                              void* d_out, int out_size, void* d_ws, size_t ws_size,
                              hipStream_t stream) {
  const int* pred = (const int*)d_in[0];
  const int* gt   = (const int*)d_in[1];
  float* out      = (float*)d_out;
  uint32_t* g_hist = (uint32_t*)d_ws;   // HSIZE u32 = 16.9 KB

  const int n  = in_sizes[0];           // 1024*1024, divisible by 4
  const int n4 = n / 4;

  iseg_init_hist<<<(HSIZE + 255) / 256, 256, 0, stream>>>(g_hist);
  iseg_hist<<<HGRID, HBLOCK, 0, stream>>>(pred, gt, g_hist, n4);
  iseg_finalize<<<1, 32, 0, stream>>>(g_hist, out);
}